// MotionSegmentationLoss_59365037965390
// MI455X (gfx1250) — compile-verified
//
#include <hip/hip_runtime.h>

typedef __attribute__((ext_vector_type(2))) float v2f;
typedef __attribute__((ext_vector_type(8))) float v8f;

#define N_PTS     16384
#define N_TILES   (N_PTS / 16)     // 1024
#define WAVES_WG  8                // 256 threads / workgroup (wave32)

// ---------------------------------------------------------------------------
// Kernel 1: normalize velocities. Writes TWO zero-padded float4 arrays:
//   U  = unit vectors            (used as the WMMA B operand)
//   Us = unit vectors * log2(e)  (used as the WMMA A operand)
// so the WMMA output is already s' = s*log2(e) and v_exp_f32 (exp2) needs
// no per-element pre-multiply. W = sum s*e^s is recovered as ln2 * sum s'*e^s.
// NOTE: Us MUST be placed directly after U in the workspace — the entropy
// kernel's final (discarded) software prefetch reads up to 2 KB past U.
// ---------------------------------------------------------------------------
__global__ void msl_normalize_kernel(const float* __restrict__ vel,
                                     float* __restrict__ U,
                                     float* __restrict__ Us) {
    int i = blockIdx.x * blockDim.x + threadIdx.x;
    if (i < N_PTS) {
        const float LOG2E = 1.44269504088896340736f;
        float x = vel[i * 3 + 0];
        float y = vel[i * 3 + 1];
        float z = vel[i * 3 + 2];
        float inv = 1.0f / (sqrtf(x * x + y * y + z * z) + 1e-6f);
        float4 o;
        o.x = x * inv; o.y = y * inv; o.z = z * inv; o.w = 0.0f;
        *(float4*)(U + (size_t)i * 4) = o;
        float4 os;
        os.x = o.x * LOG2E; os.y = o.y * LOG2E; os.z = o.z * LOG2E; os.w = 0.0f;
        *(float4*)(Us + (size_t)i * 4) = os;
    }
}

// ---------------------------------------------------------------------------
// Kernel 2: per 16-row i-tile, loop over all j-tiles:
//   S'_tile = A_scaled(16x4) x B(4x16) via v_wmma_f32_16x16x4_f32
//   accumulate Z = sum exp2(s'), W' = sum s'*exp2(s')
// entropy_row = ln2*(log2(Z) - W'/Z) - N*1e-8  (epsilon correction, 1st order)
// The B operand for iteration it+1 is prefetched via a constant-stride
// pointer bump (2048 B), so addresses fold into the load's IOFFSET and the
// L2 hit latency overlaps the WMMA + 8x v_exp_f32 of the current iteration.
// The final prefetch dangles up to 2 KB past U into Us (harmless read, discarded).
// ---------------------------------------------------------------------------
__global__ void msl_entropy_tile_kernel(const float* __restrict__ U,
                                        const float* __restrict__ Us,
                                        float* __restrict__ partials) {
    __shared__ float zsh[WAVES_WG][16];
    __shared__ float wsh[WAVES_WG][16];
    __shared__ float ent[16];

    const int lane = threadIdx.x & 31;
    const int wave = threadIdx.x >> 5;
    const int m    = lane & 15;       // row/col within tile
    const int hi   = lane >> 4;       // half-wave selector
    const int i0   = blockIdx.x << 4;
    const int loff = m * 4 + hi * 2;  // per-lane float offset within a tile

    // A operand (loop-invariant, pre-scaled by log2e): lanes 0-15 hold
    // (x,y)=K0,K1 of row i0+m; lanes 16-31 hold (z,0)=K2,K3.
    v2f a = *(const v2f*)(Us + (size_t)(i0 + m) * 4 + hi * 2);

    float zacc[8], wacc[8];
#pragma unroll
    for (int v = 0; v < 8; ++v) { zacc[v] = 0.0f; wacc[v] = 0.0f; }

    const int    steps  = N_TILES / WAVES_WG;   // 128, uniform trip count
    const size_t stride = (size_t)WAVES_WG * 64; // floats between j-tiles of a wave

    const float* bptr = U + (size_t)(wave << 6) + loff;
    v2f b = *(const v2f*)bptr;

#pragma unroll 2
    for (int it = 0; it < steps; ++it) {
        // prefetch next j-tile's B operand (constant-stride pointer bump)
        bptr += stride;
        v2f bn = *(const v2f*)bptr;

        v8f c = {0.0f, 0.0f, 0.0f, 0.0f, 0.0f, 0.0f, 0.0f, 0.0f};
        c = __builtin_amdgcn_wmma_f32_16x16x4_f32(
                /*neg_a=*/false, a, /*neg_b=*/false, b,
                /*c_mod=*/(short)0, c, /*reuse_a=*/false, /*reuse_b=*/false);

#pragma unroll
        for (int v = 0; v < 8; ++v) {
            float sp = c[v];                            // s * log2(e)
            float e  = __builtin_amdgcn_exp2f(sp);      // v_exp_f32 == e^s
            zacc[v] += e;
            wacc[v] += sp * e;                          // accumulates W/ln2
        }

        b = bn;
    }

    // Reduce across the 16 lanes of each half-wave (offsets < 16 keep the
    // two halves independent). Row M = v + 8*hi.
#pragma unroll
    for (int v = 0; v < 8; ++v) {
#pragma unroll
        for (int off = 8; off >= 1; off >>= 1) {
            zacc[v] += __shfl_xor(zacc[v], off, 32);
            wacc[v] += __shfl_xor(wacc[v], off, 32);
        }
    }
    if (m == 0) {
#pragma unroll
        for (int v = 0; v < 8; ++v) {
            zsh[wave][hi * 8 + v] = zacc[v];
            wsh[wave][hi * 8 + v] = wacc[v];
        }
    }
    __syncthreads();

    if (threadIdx.x < 16) {
        float Z = 0.0f, Wp = 0.0f;
        for (int w2 = 0; w2 < WAVES_WG; ++w2) {
            Z  += zsh[w2][threadIdx.x];
            Wp += wsh[w2][threadIdx.x];
        }
        const float LN2 = 0.6931471805599453f;
        // entropy = ln2*(log2(Z) - W'/Z) - N*1e-8 (epsilon correction)
        float e_i = LN2 * (__builtin_amdgcn_logf(Z) - Wp / Z) - 1.6384e-4f;
        ent[threadIdx.x] = e_i;
    }
    __syncthreads();

    if (threadIdx.x == 0) {
        float s = 0.0f;
#pragma unroll
        for (int r = 0; r < 16; ++r) s += ent[r];
        partials[blockIdx.x] = s;
    }
}

// ---------------------------------------------------------------------------
// Kernel 3: deterministic fixed-order reduction of 1024 partials -> mean.
// ---------------------------------------------------------------------------
__global__ void msl_final_reduce_kernel(const float* __restrict__ partials,
                                        float* __restrict__ out) {
    __shared__ float sm[256];
    int t = threadIdx.x;
    float s = partials[t] + partials[t + 256] +
              partials[t + 512] + partials[t + 768];
    sm[t] = s;
    __syncthreads();
    for (int off = 128; off > 0; off >>= 1) {
        if (t < off) sm[t] += sm[t + off];
        __syncthreads();
    }
    if (t == 0) out[0] = sm[0] / (float)N_PTS;
}

// ---------------------------------------------------------------------------
extern "C" void kernel_launch(void* const* d_in, const int* in_sizes, int n_in,
                              void* d_out, int out_size, void* d_ws, size_t ws_size,
                              hipStream_t stream) {
    (void)in_sizes; (void)n_in; (void)out_size; (void)ws_size;

    const float* vel = (const float*)d_in[0];   // velocities (N,3) fp32
    // d_in[1] (positions) is unused by the reference math.

    float* Uf       = (float*)d_ws;                      // 16384*4 floats (256 KB)
    float* Usf      = Uf + (size_t)N_PTS * 4;            // 16384*4 floats (256 KB)
    float* partials = Usf + (size_t)N_PTS * 4;           // 1024 floats
    float* out      = (float*)d_out;

    msl_normalize_kernel<<<(N_PTS + 255) / 256, 256, 0, stream>>>(vel, Uf, Usf);
    msl_entropy_tile_kernel<<<N_TILES, 32 * WAVES_WG, 0, stream>>>(Uf, Usf, partials);
    msl_final_reduce_kernel<<<1, 256, 0, stream>>>(partials, out);
}